// NTMHead_72593537237206
// MI455X (gfx1250) — compile-verified
//
#include <hip/hip_runtime.h>
#include <hip/hip_bf16.h>
#include <math.h>

#define EPS 1e-8f

typedef float v2f __attribute__((ext_vector_type(2)));
typedef float v8f __attribute__((ext_vector_type(8)));

// ---------------------------------------------------------------------------
// Kernel 1: out = controller_out(512x1024) @ W_fc(1024x134) + b_fc
// fp32 WMMA (V_WMMA_F32_16X16X4_F32), one wave per 16x16 output tile.
// Output written to workspace with padded stride 144.
// ---------------------------------------------------------------------------
__global__ __launch_bounds__(32)
void ntm_fc_wmma(const float* __restrict__ A,     // 512 x 1024
                 const float* __restrict__ Bm,    // 1024 x 134
                 const float* __restrict__ bias,  // 134
                 float* __restrict__ outw)        // 512 x 144 (padded)
{
    const int tileN = blockIdx.x;        // 0..8  (cols, 16 each, 144 padded)
    const int tileM = blockIdx.y;        // 0..31 (rows, 16 each)
    const int lane  = threadIdx.x;       // 0..31 (wave32)
    const int r     = lane & 15;
    const int half  = lane >> 4;

    const int row0 = tileM * 16;
    const int col0 = tileN * 16;
    // Clamp W_fc column for padded tail tile (cols >= 134 are never read back)
    const int colB = (col0 + r < 134) ? (col0 + r) : 133;

    v8f c = {};
    for (int k = 0; k < 1024; k += 4) {
        const int ka = k + half * 2;   // K pair handled by this lane-half
        v2f a, b;
        // A 16x4 fragment: v0 = {K0 | K2}, v1 = {K1 | K3} across lane halves
        a.x = A[(row0 + r) * 1024 + ka];
        a.y = A[(row0 + r) * 1024 + ka + 1];
        // B 4x16 fragment: v0 = rows {K0 | K2}, v1 = rows {K1 | K3}, col = r
        b.x = Bm[(size_t)ka * 134 + colB];
        b.y = Bm[(size_t)(ka + 1) * 134 + colB];
        c = __builtin_amdgcn_wmma_f32_16x16x4_f32(
                /*neg_a=*/false, a, /*neg_b=*/false, b,
                /*c_mod=*/(short)0, c, /*reuse_a=*/false, /*reuse_b=*/false);
    }

    const int col = col0 + r;
    const float bv = (col < 134) ? bias[col] : 0.0f;
    // C/D layout: VGPR j -> row j (lanes 0-15) / row j+8 (lanes 16-31), N = lane&15
    #pragma unroll
    for (int j = 0; j < 8; ++j) {
        const int row = row0 + j + half * 8;
        outw[(size_t)row * 144 + col] = c[j] + bv;
    }
}

// ---------------------------------------------------------------------------
// Kernel 2: fused NTM addressing. One 256-thread workgroup per batch b.
// mem[b] (256 KB) is cached in CDNA5's 320 KB LDS: one HBM pass total.
// ---------------------------------------------------------------------------
__device__ inline float softplusf(float x) {
    return (x > 20.0f) ? x : log1pf(expf(x));
}

#define SM_MEM    0                      // 512*128 floats (mem[b] tile)
#define SM_K      (512 * 128)            // 128
#define SM_SIM    (SM_K + 128)           // 512 (reused later for final w)
#define SM_WG     (SM_SIM + 512)         // 512
#define SM_RED    (SM_WG + 512)          // 256
#define SM_SCAL   (SM_RED + 256)         // 8
#define SM_PART   (SM_SCAL + 8)          // 1024 (8 waves x 128 cols, 16B aligned)
#define SM_FLOATS (SM_PART + 1024)

__global__ __launch_bounds__(256)
void ntm_fused(const float* __restrict__ outw,   // 512 x 144 (padded FC out)
               const float* __restrict__ prev_w, // 512 x 512
               const float* __restrict__ mem,    // 512 x 512 x 128
               float* __restrict__ out)          // [512*128 read_vec | 512*512 w]
{
    extern __shared__ float smem[];
    float* mem_s = smem + SM_MEM;
    float* k_s   = smem + SM_K;
    float* sim_s = smem + SM_SIM;
    float* wg_s  = smem + SM_WG;
    float* red   = smem + SM_RED;
    float* scal  = smem + SM_SCAL;
    float* part  = smem + SM_PART;

    const int b    = blockIdx.x;
    const int tid  = threadIdx.x;         // 0..255
    const int wave = tid >> 5;            // 0..7
    const int lane = tid & 31;            // 0..31

    // ---- Phase A: key vector, norm_k, gate scalars --------------------------
    const float ov = (tid < 134) ? outw[(size_t)b * 144 + tid] : 0.0f;
    if (tid < 128) k_s[tid] = ov;
    if (tid >= 128 && tid < 134) scal[tid - 128] = ov;   // raw beta,g,s0,s1,s2,gamma
    red[tid] = (tid < 128) ? ov * ov : 0.0f;
    __syncthreads();
    for (int s = 128; s > 0; s >>= 1) {
        if (tid < s) red[tid] += red[tid + s];
        __syncthreads();
    }
    if (tid == 0) {
        const float norm_k = sqrtf(red[0]) + EPS;
        const float beta  = softplusf(scal[0]);
        const float g     = 1.0f / (1.0f + expf(-scal[1]));
        const float m3    = fmaxf(scal[2], fmaxf(scal[3], scal[4]));
        const float e0 = expf(scal[2] - m3), e1 = expf(scal[3] - m3), e2 = expf(scal[4] - m3);
        const float esum = e0 + e1 + e2;
        const float gamma = 1.0f + softplusf(scal[5]);
        scal[0] = beta; scal[1] = g;
        scal[2] = e0 / esum; scal[3] = e1 / esum; scal[4] = e2 / esum;
        scal[5] = gamma; scal[6] = norm_k;
    }
    __syncthreads();

    // ---- Phase B: stream mem[b] HBM->LDS (b128), cosine similarity ----------
    {
        const size_t base = ((size_t)b * 512) * 128;
        const float4 kv = ((const float4*)k_s)[lane];     // k[4*lane .. 4*lane+3]
        #pragma unroll 2
        for (int n = wave * 64; n < wave * 64 + 64; ++n) {
            const float4 v = ((const float4*)(mem + base + (size_t)n * 128))[lane];
            ((float4*)(mem_s + n * 128))[lane] = v;        // ds_store_b128
            float dot = v.x * kv.x + v.y * kv.y + v.z * kv.z + v.w * kv.w;
            float sq  = v.x * v.x + v.y * v.y + v.z * v.z + v.w * v.w;
            #pragma unroll
            for (int off = 16; off > 0; off >>= 1) {
                dot += __shfl_xor(dot, off, 32);
                sq  += __shfl_xor(sq,  off, 32);
            }
            if (lane == 0)
                sim_s[n] = dot / (scal[6] * (sqrtf(sq) + EPS));
        }
    }
    __syncthreads();

    // ---- Phase C: softmax, interpolate, shift, sharpen ----------------------
    const float beta = scal[0], g = scal[1];
    const float s0 = scal[2], s1 = scal[3], s2 = scal[4];
    const float gamma = scal[5];
    const int n0 = tid, n1 = tid + 256;

    const float x0 = beta * sim_s[n0];
    const float x1 = beta * sim_s[n1];
    red[tid] = fmaxf(x0, x1);
    __syncthreads();
    for (int s = 128; s > 0; s >>= 1) {
        if (tid < s) red[tid] = fmaxf(red[tid], red[tid + s]);
        __syncthreads();
    }
    const float mx = red[0];
    __syncthreads();
    const float e0 = expf(x0 - mx), e1 = expf(x1 - mx);
    red[tid] = e0 + e1;
    __syncthreads();
    for (int s = 128; s > 0; s >>= 1) {
        if (tid < s) red[tid] += red[tid + s];
        __syncthreads();
    }
    const float esum = red[0];
    __syncthreads();

    const float wg0 = g * (e0 / esum) + (1.0f - g) * prev_w[(size_t)b * 512 + n0];
    const float wg1 = g * (e1 / esum) + (1.0f - g) * prev_w[(size_t)b * 512 + n1];
    wg_s[n0] = wg0;
    wg_s[n1] = wg1;
    __syncthreads();

    // circular 3-tap: w_s[n] = s0*w_g[n-1] + s1*w_g[n] + s2*w_g[n+1]
    const float a0 = s0 * wg_s[(n0 + 511) & 511] + s1 * wg_s[n0] + s2 * wg_s[(n0 + 1) & 511];
    const float a1 = s0 * wg_s[(n1 + 511) & 511] + s1 * wg_s[n1] + s2 * wg_s[(n1 + 1) & 511];
    const float wp0 = powf(a0, gamma);
    const float wp1 = powf(a1, gamma);
    red[tid] = wp0 + wp1;
    __syncthreads();
    for (int s = 128; s > 0; s >>= 1) {
        if (tid < s) red[tid] += red[tid + s];
        __syncthreads();
    }
    const float denom = red[0] + EPS;
    __syncthreads();

    const float wf0 = wp0 / denom;
    const float wf1 = wp1 / denom;
    sim_s[n0] = wf0;                      // reuse as final w
    sim_s[n1] = wf1;
    out[512 * 128 + (size_t)b * 512 + n0] = wf0;
    out[512 * 128 + (size_t)b * 512 + n1] = wf1;
    __syncthreads();

    // ---- Phase D: read_vec = w @ mem[b], entirely from LDS (b128 reads) -----
    {
        // wave `wave` owns rows [wave*64, wave*64+64); lane owns cols 4*lane..+3
        float4 acc = make_float4(0.f, 0.f, 0.f, 0.f);
        for (int n = wave * 64; n < wave * 64 + 64; ++n) {
            const float wv = sim_s[n];
            const float4 m4 = ((const float4*)(mem_s + n * 128))[lane];
            acc.x += wv * m4.x;
            acc.y += wv * m4.y;
            acc.z += wv * m4.z;
            acc.w += wv * m4.w;
        }
        ((float4*)(part + wave * 128))[lane] = acc;
        __syncthreads();
        if (tid < 128) {
            float rv = 0.f;
            #pragma unroll
            for (int w = 0; w < 8; ++w) rv += part[w * 128 + tid];
            out[(size_t)b * 128 + tid] = rv;
        }
    }
}

// ---------------------------------------------------------------------------
extern "C" void kernel_launch(void* const* d_in, const int* in_sizes, int n_in,
                              void* d_out, int out_size, void* d_ws, size_t ws_size,
                              hipStream_t stream) {
    const float* controller_out = (const float*)d_in[0]; // 512 x 1024
    const float* prev_w         = (const float*)d_in[1]; // 512 x 512
    const float* memory_state   = (const float*)d_in[2]; // 512 x 512 x 128
    const float* W_fc           = (const float*)d_in[3]; // 1024 x 134
    const float* b_fc           = (const float*)d_in[4]; // 134
    float* out = (float*)d_out;                          // 512*128 + 512*512
    float* fc_ws = (float*)d_ws;                         // 512 x 144 fp32

    // FC GEMM via fp32 WMMA: 9 col-tiles x 32 row-tiles, one wave per tile
    dim3 g1(9, 32);
    ntm_fc_wmma<<<g1, 32, 0, stream>>>(controller_out, W_fc, b_fc, fc_ws);

    // Fused NTM head: one workgroup per batch row, ~272 KB LDS (CDNA5 320 KB)
    const size_t smem_bytes = (size_t)SM_FLOATS * sizeof(float);
    ntm_fused<<<512, 256, smem_bytes, stream>>>(fc_ws, prev_w, memory_state, out);
}